// HierarchPostProcessor_49469433315560
// MI455X (gfx1250) — compile-verified
//
#include <hip/hip_runtime.h>
#include <hip/hip_bf16.h>
#include <stdint.h>

// Problem constants
#define BB   8
#define RR   32768
#define NN   512
#define CC   151
#define R3   98304      // 3*RR
#define PAD  131072     // next pow2 >= R3

typedef __attribute__((ext_vector_type(8))) float v8f;
typedef __attribute__((ext_vector_type(2))) float v2f;

__constant__ int GEO_T[15] = {1, 2, 3, 4, 5, 6, 8, 10, 22, 23, 29, 31, 32, 33, 43};
__constant__ int POS_T[11] = {9, 16, 17, 20, 27, 30, 36, 42, 48, 49, 50};
__constant__ int SEM_T[24] = {7, 11, 12, 13, 14, 15, 18, 19, 21, 24, 25, 26,
                              28, 34, 35, 37, 38, 39, 40, 41, 44, 45, 46, 47};

// ---------- CDNA5 async LDS<->global helpers (inline asm; gfx1250 only) ----------
__device__ __forceinline__ unsigned lds_off32(const void* p) {
  // LDS flat addr = {SHARED_BASE, offset[31:0]} -> low 32 bits are the LDS byte offset
  return (unsigned)(unsigned long long)(uintptr_t)p;
}
__device__ __forceinline__ void async_load_b128(unsigned lds, unsigned long long ga) {
  asm volatile("global_load_async_to_lds_b128 %0, %1, off" :: "v"(lds), "v"(ga) : "memory");
}
__device__ __forceinline__ void async_store_b128(unsigned long long ga, unsigned lds) {
  asm volatile("global_store_async_from_lds_b128 %0, %1, off" :: "v"(ga), "v"(lds) : "memory");
}
__device__ __forceinline__ void wait_async0() {
  asm volatile("s_wait_asynccnt 0" ::: "memory");
}
__device__ __forceinline__ void wait_ds0() {
  asm volatile("s_wait_dscnt 0" ::: "memory");
}

// Map float -> u32 so that ascending u64 sort == descending float, stable by index
__device__ __forceinline__ unsigned long long pack_key(float t, unsigned idx) {
  unsigned u = __float_as_uint(t);
  unsigned s = (u & 0x80000000u) ? ~u : (u | 0x80000000u);  // ascending order map
  unsigned d = ~s;                                          // descending
  return ((unsigned long long)d << 32) | (unsigned long long)idx;
}

// ============================================================================
// Kernel 1: object branch. 1 wave per 16 rows. Rows staged into LDS once via
// async-to-LDS; softmax denominator via WMMA f32 16x16x4 (B = ones => D rows
// are row-sums, accumulated over K chunks). Only the last K-chunk is masked,
// via cndmask (no branches in the hot loop).
// ============================================================================
__global__ __launch_bounds__(32) void obj_kernel(const float* __restrict__ logits,
                                                 float* __restrict__ out_pred,
                                                 float* __restrict__ out_scores,
                                                 float* __restrict__ ws_scores) {
  __shared__ __align__(16) float xs[16 * CC + 8];  // 9664B staged + pad for x[151] read
  __shared__ float smax[16];
  __shared__ float sm1[16];
  __shared__ int   sam[16];
  __shared__ float sS[16];

  const int lane = threadIdx.x;
  const int row0 = blockIdx.x * 16;  // 256 blocks * 16 rows = 4096 = B*N

  // --- async stage: 16 rows * 151 f32 = 9664 B = 604 b128 chunks (16B aligned) ---
  {
    const unsigned l0 = lds_off32(xs);
    unsigned long long ga =
        (unsigned long long)(uintptr_t)(logits + (size_t)row0 * CC);
    for (int i = lane; i < 604; i += 32)
      async_load_b128(l0 + (unsigned)i * 16u, ga + (unsigned long long)i * 16u);
  }
  wait_async0();

  // --- Phase 1: per-row global max (all 151) and max/argmax over classes >= 1 ---
  for (int jr = 0; jr < 16; ++jr) {
    const float* x = &xs[jr * CC];
    float m = -3.0e38f, m1 = -3.0e38f;
    int am = 0x7fffffff;
    for (int c = lane; c < CC; c += 32) {
      float v = x[c];
      m = fmaxf(m, v);
      if (c >= 1 && v > m1) { m1 = v; am = c; }
    }
    for (int off = 16; off > 0; off >>= 1) {
      float om = __shfl_xor(m, off);
      m = fmaxf(m, om);
      float o1 = __shfl_xor(m1, off);
      int   oa = __shfl_xor(am, off);
      if (o1 > m1 || (o1 == m1 && oa < am)) { m1 = o1; am = oa; }
    }
    if (lane == 0) { smax[jr] = m; sm1[jr] = m1; sam[jr] = am; }
  }
  __syncthreads();

  // --- Phase 2: WMMA row-sum of exp(x - m), fed from LDS ---
  // A 16x4 f32 layout: lanes 0-15 hold M=lane (K=0 in v0, K=1 in v1);
  // lanes 16-31 hold M=lane-16 (K=2 in v0, K=3 in v1).
  const int row  = lane & 15;
  const int koff = (lane >> 4) * 2;
  const float m  = smax[row];
  const float* x = &xs[row * CC];

  v8f acc = {0.f, 0.f, 0.f, 0.f, 0.f, 0.f, 0.f, 0.f};
  v2f bones; bones.x = 1.0f; bones.y = 1.0f;  // all-ones B => D[m][n] = sum_k A[m][k]

  // Chunks k0 = 0..144: max index = 144 + 2 + 1 = 147 < 151 -> no masking at all.
#pragma unroll 4
  for (int k0 = 0; k0 < 148; k0 += 4) {
    v2f a;
    a.x = __expf(x[k0 + koff]     - m);
    a.y = __expf(x[k0 + koff + 1] - m);
    acc = __builtin_amdgcn_wmma_f32_16x16x4_f32(false, a, false, bones,
                                                (short)0, acc, false, false);
  }
  // Final chunk k0 = 148: c0 in {148,150}; only element 151 is out of range.
  {
    const int c0 = 148 + koff;
    v2f a;
    a.x = __expf(x[c0] - m);                       // c0 <= 150, in range
    float e1 = __expf(x[c0 + 1] - m);              // may read padded slot; selected away
    a.y = (c0 + 1 < CC) ? e1 : 0.0f;               // v_cndmask, no branch
    acc = __builtin_amdgcn_wmma_f32_16x16x4_f32(false, a, false, bones,
                                                (short)0, acc, false, false);
  }

  // D layout: vgpr j of lane L holds D[j + 8*(L>>4)][L&15]; take column N=0.
  if ((lane & 15) == 0) {
    int mh = (lane >> 4) * 8;
#pragma unroll
    for (int jj = 0; jj < 8; ++jj) sS[mh + jj] = acc[jj];
  }
  __syncthreads();

  if (lane < 16) {
    int r = row0 + lane;
    float S = sS[lane];
    float score = __expf(sm1[lane] - smax[lane]) / S;  // max softmax prob over c>=1
    out_pred[r]   = (float)sam[lane];                  // label = argmax(prob[1:]) + 1
    out_scores[r] = score;
    ws_scores[r]  = score;
  }
}

// ============================================================================
// Kernel 2: relation branch. Each wave stages 32 rows of rel1/2/3 into LDS via
// async-to-LDS b128, computes exps + per-hierarchy max/argmax, streams the
// 50-wide full_prob rows back out via async-from-LDS b128, and writes sort
// keys + labels.
// ============================================================================
__global__ __launch_bounds__(256) void rel_kernel(const float* __restrict__ rel1,
                                                  const float* __restrict__ rel2,
                                                  const float* __restrict__ rel3,
                                                  const int* __restrict__ pairs,
                                                  const float* __restrict__ ws_objs,
                                                  float* __restrict__ ws_full,
                                                  unsigned long long* __restrict__ ws_keys,
                                                  float* __restrict__ ws_labels) {
  __shared__ __align__(16) float ldsIn[8 * 1600];  // 8 waves * 32 rows * 50 floats

  const int wave = threadIdx.x >> 5;
  const int lane = threadIdx.x & 31;
  const int g0   = blockIdx.x * 256;       // 1024 blocks * 256 rows = B*R
  const int b    = g0 / RR;
  const int rbase = (g0 % RR) + wave * 32; // this wave's 32 rows

  float* inBase = &ldsIn[wave * 1600];
  const unsigned l0 = lds_off32(inBase);

  // --- async stage: [32x15 | 32x11 | 32x24] = 1920 + 1408 + 3072 bytes ---
  {
    unsigned long long ga1 =
        (unsigned long long)(uintptr_t)(rel1 + ((size_t)b * RR + rbase) * 15);
    for (int i = lane; i < 120; i += 32)
      async_load_b128(l0 + (unsigned)i * 16u, ga1 + (unsigned long long)i * 16u);
    unsigned long long ga2 =
        (unsigned long long)(uintptr_t)(rel2 + ((size_t)b * RR + rbase) * 11);
    for (int i = lane; i < 88; i += 32)
      async_load_b128(l0 + 1920u + (unsigned)i * 16u, ga2 + (unsigned long long)i * 16u);
    unsigned long long ga3 =
        (unsigned long long)(uintptr_t)(rel3 + ((size_t)b * RR + rbase) * 24);
    for (int i = lane; i < 192; i += 32)
      async_load_b128(l0 + 3328u + (unsigned)i * 16u, ga3 + (unsigned long long)i * 16u);
  }
  wait_async0();

  // --- pull this lane's row into registers (segmented layout) ---
  const float* i1 = inBase + lane * 15;
  const float* i2 = inBase + 480 + lane * 11;
  const float* i3 = inBase + 832 + lane * 24;
  float v[50];
#pragma unroll
  for (int i = 0; i < 15; ++i) v[i] = i1[i];
#pragma unroll
  for (int i = 0; i < 11; ++i) v[15 + i] = i2[i];
#pragma unroll
  for (int i = 0; i < 24; ++i) v[26 + i] = i3[i];

  // raw max/argmax per hierarchy (argmax(exp) == argmax(raw); max(exp) = exp(max))
  float m1 = v[0];  int a1 = 0;
  float m2 = v[15]; int a2 = 0;
  float m3 = v[26]; int a3 = 0;
#pragma unroll
  for (int i = 1; i < 15; ++i) if (v[i]      > m1) { m1 = v[i];      a1 = i; }
#pragma unroll
  for (int i = 1; i < 11; ++i) if (v[15 + i] > m2) { m2 = v[15 + i]; a2 = i; }
#pragma unroll
  for (int i = 1; i < 24; ++i) if (v[26 + i] > m3) { m3 = v[26 + i]; a3 = i; }
  const float s1 = __expf(m1), s2 = __expf(m2), s3 = __expf(m3);

  // exp in place, then write interleaved 50-wide row back to LDS (wave lockstep:
  // all DS loads above precede these DS stores in the wave's in-order DS stream)
  float* o = inBase + lane * 50;
#pragma unroll
  for (int i = 0; i < 50; ++i) o[i] = __expf(v[i]);
  wait_ds0();  // LDS writes complete before the async engine reads them

  // --- async stream full_prob (32 rows * 200B = 400 b128 chunks) ---
  {
    unsigned long long go =
        (unsigned long long)(uintptr_t)(ws_full + ((size_t)b * RR + rbase) * 50);
    for (int i = lane; i < 400; i += 32)
      async_store_b128(go + (unsigned long long)i * 16u, l0 + (unsigned)i * 16u);
  }

  // --- triple scores -> packed sort keys; labels ---
  const int r  = rbase + lane;
  const int p0 = pairs[((size_t)b * RR + r) * 2 + 0];
  const int p1 = pairs[((size_t)b * RR + r) * 2 + 1];
  const float oo = ws_objs[b * NN + p0] * ws_objs[b * NN + p1];

  unsigned long long* kb = ws_keys + (size_t)b * PAD;
  float* lb = ws_labels + (size_t)b * R3;
  kb[r]          = pack_key(s1 * oo, (unsigned)r);
  kb[RR + r]     = pack_key(s2 * oo, (unsigned)(RR + r));
  kb[2 * RR + r] = pack_key(s3 * oo, (unsigned)(2 * RR + r));
  lb[r]          = (float)GEO_T[a1];
  lb[RR + r]     = (float)POS_T[a2];
  lb[2 * RR + r] = (float)SEM_T[a3];
  // s_endpgm performs an implicit wait-idle, covering outstanding async stores
}

// ============================================================================
// Kernel 3: pad each batch's key region [R3, PAD) with +inf keys
// ============================================================================
__global__ __launch_bounds__(256) void pad_kernel(unsigned long long* __restrict__ keys) {
  int t = blockIdx.x * blockDim.x + threadIdx.x;  // 8 * 32768 threads
  int b = t >> 15;
  int i = t & 32767;
  keys[(size_t)b * PAD + R3 + i] = ~0ull;
}

// ============================================================================
// Kernel 4a: global bitonic stage (j > 2048)
// ============================================================================
__global__ __launch_bounds__(256) void bitonic_global(unsigned long long* __restrict__ keys,
                                                      int j, int k) {
  int t = blockIdx.x * blockDim.x + threadIdx.x;  // B*PAD threads
  int b = t >> 17;                                 // PAD = 2^17
  int i = t & (PAD - 1);
  int ixj = i ^ j;
  if (ixj > i) {
    unsigned long long* base = keys + (size_t)b * PAD;
    unsigned long long a = base[i];
    unsigned long long c = base[ixj];
    bool up = ((i & k) == 0);
    if ((a > c) == up) { base[i] = c; base[ixj] = a; }
  }
}

// ============================================================================
// Kernel 4b: LDS-fused bitonic stages j = jstart..1 over 4096-element tiles
// ============================================================================
__global__ __launch_bounds__(256) void bitonic_fused(unsigned long long* __restrict__ keys,
                                                     int k, int jstart) {
  __shared__ unsigned long long tile[4096];  // 32 KB
  const int b   = blockIdx.x >> 5;           // PAD/4096 = 32 tiles per batch
  const int seg = blockIdx.x & 31;
  unsigned long long* base = keys + (size_t)b * PAD + (size_t)seg * 4096;
  const int gbase = seg * 4096;

  for (int x = threadIdx.x; x < 4096; x += 256) tile[x] = base[x];
  __syncthreads();

  for (int j = jstart; j >= 1; j >>= 1) {
    for (int p = threadIdx.x; p < 2048; p += 256) {
      int i   = ((p & ~(j - 1)) << 1) | (p & (j - 1));
      int ixj = i | j;
      bool up = (((gbase + i) & k) == 0);
      unsigned long long a = tile[i];
      unsigned long long c = tile[ixj];
      if ((a > c) == up) { tile[i] = c; tile[ixj] = a; }
    }
    __syncthreads();
  }

  for (int x = threadIdx.x; x < 4096; x += 256) base[x] = tile[x];
}

// ============================================================================
// Kernel 5: gather sorted outputs
// ============================================================================
__global__ __launch_bounds__(256) void gather_kernel(const unsigned long long* __restrict__ keys,
                                                     const float* __restrict__ ws_full,
                                                     const float* __restrict__ ws_labels,
                                                     const int* __restrict__ pairs,
                                                     float* __restrict__ out_pairs,
                                                     float* __restrict__ out_probs,
                                                     float* __restrict__ out_labels) {
  const int wave = threadIdx.x >> 5;
  const int lane = threadIdx.x & 31;
  const int g0 = blockIdx.x * 256 + wave * 32;  // 3072 blocks * 256 = B*R3
  const int b  = g0 / R3;
  const int jbase = g0 % R3;
  const int j = jbase + lane;

  const unsigned idx = (unsigned)keys[(size_t)b * PAD + j];  // low 32 bits = cat index
  const int r = (int)(idx & (RR - 1));                       // RR is a power of two

  out_labels[(size_t)b * R3 + j] = ws_labels[(size_t)b * R3 + idx];
  out_pairs[((size_t)b * R3 + j) * 2 + 0] = (float)pairs[((size_t)b * RR + r) * 2 + 0];
  out_pairs[((size_t)b * R3 + j) * 2 + 1] = (float)pairs[((size_t)b * RR + r) * 2 + 1];

  // Cooperative, coalesced copy of the 50-wide prob row (8B-aligned -> float2)
  for (int l = 0; l < 32; ++l) {
    int rr = __shfl(r, l);
    const float2* src = (const float2*)(ws_full + ((size_t)b * RR + rr) * 50);
    float2* dst = (float2*)(out_probs + ((size_t)b * R3 + jbase + l) * 50);
    if (lane < 25) dst[lane] = src[lane];
  }
}

// ============================================================================
// Host launcher
// ============================================================================
extern "C" void kernel_launch(void* const* d_in, const int* in_sizes, int n_in,
                              void* d_out, int out_size, void* d_ws, size_t ws_size,
                              hipStream_t stream) {
  (void)in_sizes; (void)n_in; (void)out_size; (void)ws_size;

  const float* rel1       = (const float*)d_in[0];
  const float* rel2       = (const float*)d_in[1];
  const float* rel3       = (const float*)d_in[2];
  /* d_in[3] = super_rel_probs: unused by the reference */
  const float* obj_logits = (const float*)d_in[4];
  const int*   pairs      = (const int*)d_in[5];

  float* out        = (float*)d_out;
  float* out_pred   = out;             // [B,N]        4096
  float* out_scores = out + 4096;      // [B,N]        4096
  float* out_pairs  = out + 8192;      // [B,3R,2]     1572864
  float* out_probs  = out + 1581056;   // [B,3R,50]    39321600
  float* out_labels = out + 40902656;  // [B,3R]       786432

  char* ws = (char*)d_ws;
  float*              ws_full   = (float*)(ws);                         // 52,428,800 B
  unsigned long long* ws_keys   = (unsigned long long*)(ws + 52428800); //  8,388,608 B
  float*              ws_labels = (float*)(ws + 60817408);              //  3,145,728 B
  float*              ws_objs   = (float*)(ws + 63963136);              //     16,384 B

  obj_kernel<<<256, 32, 0, stream>>>(obj_logits, out_pred, out_scores, ws_objs);
  rel_kernel<<<1024, 256, 0, stream>>>(rel1, rel2, rel3, pairs, ws_objs,
                                       ws_full, ws_keys, ws_labels);
  pad_kernel<<<1024, 256, 0, stream>>>(ws_keys);

  for (int k = 2; k <= PAD; k <<= 1) {
    int j = k >> 1;
    for (; j > 2048; j >>= 1)
      bitonic_global<<<(BB * PAD) / 256, 256, 0, stream>>>(ws_keys, j, k);
    bitonic_fused<<<(BB * PAD) / 4096, 256, 0, stream>>>(ws_keys, k, j);
  }

  gather_kernel<<<(BB * R3) / 256, 256, 0, stream>>>(ws_keys, ws_full, ws_labels,
                                                     pairs, out_pairs, out_probs,
                                                     out_labels);
}